// EPGSimulation_63694365000168
// MI455X (gfx1250) — compile-verified
//
#include <hip/hip_runtime.h>
#include <math.h>

#define N_PULSES 200
#define BATCH 2048
#define N_STATES 64
#define WAVES_PER_BLOCK 8
#define BLOCK_SZ (WAVES_PER_BLOCK * 32)

typedef float v2f __attribute__((ext_vector_type(2)));
typedef float v4f __attribute__((ext_vector_type(4)));

__global__ __launch_bounds__(BLOCK_SZ)
void epg_sim_kernel(const float* __restrict__ flip_angles,
                    const float* __restrict__ phases,
                    const float* __restrict__ T1,
                    const float* __restrict__ T2,
                    const int*   __restrict__ TRp,
                    const float* __restrict__ B0,
                    const float* __restrict__ B1,
                    float* __restrict__ out)
{
    // Block-wide per-pulse phase factors: {cos p, sin p, cos 2p, sin 2p}
    __shared__ v4f s_ph[N_PULSES];
    // Per-wave RF table: {sin(flip*B1), cos(flip*B1)} -- wave-uniform values,
    // computed cooperatively (lanes split pulses) instead of 200x per lane.
    __shared__ v2f s_rf[WAVES_PER_BLOCK][N_PULSES];

    const int lane = threadIdx.x & 31;
    const int wave = threadIdx.x >> 5;
    const int b    = blockIdx.x * WAVES_PER_BLOCK + wave;

    for (int p = threadIdx.x; p < N_PULSES; p += BLOCK_SZ) {
        float s, c;
        sincosf(phases[p], &s, &c);
        v4f t = {c, s, c * c - s * s, 2.0f * c * s};
        s_ph[p] = t;
    }

    const float b1 = B1[b];
    for (int p = lane; p < N_PULSES; p += 32) {   // ~7 sincos per lane
        float sa, ca;
        sincosf(flip_angles[p] * b1, &sa, &ca);
        v2f t = {sa, ca};
        s_rf[wave][p] = t;
    }
    __syncthreads();

    // Per-batch-element constants (wave-uniform).
    const float TR  = (float)(*TRp);
    const float E1  = expf(-TR / T1[b]);
    const float E2  = expf(-TR / T2[b]);
    const float b1rec = 1.0f - E1;
    const float rec0  = (lane == 0) ? b1rec : 0.0f;
    const float phi = 2.0f * 3.14159265358979323846f * B0[b] * TR * 0.001f;
    float b0i, b0r;
    sincosf(phi, &b0i, &b0r);          // b0_phase = b0r + i*b0i

    // Lane L owns states 2L (j=0) and 2L+1 (j=1); all state lives in VGPRs.
    float fpr[2] = {0.0f, 0.0f}, fpi[2] = {0.0f, 0.0f};
    float fmr[2] = {0.0f, 0.0f}, fmi[2] = {0.0f, 0.0f};
    float z[2];
    z[0] = (lane == 0) ? 1.0f : 0.0f;
    z[1] = 0.0f;

    const size_t CPLX = (size_t)N_PULSES * BATCH * N_STATES * 2;
    float* __restrict__ outFp = out;
    float* __restrict__ outFm = out + CPLX;
    float* __restrict__ outZ  = out + 2 * CPLX;

    for (int p = 0; p < N_PULSES; ++p) {
        // Branch-free, transcendental-free hot loop: 2 LDS broadcast loads.
        const v4f ph = s_ph[p];
        const float er = ph.x, ei = ph.y, e2r = ph.z, e2i = ph.w;
        const v2f rf = s_rf[wave][p];
        const float sa = rf.x, ca = rf.y;
        // Half-angle identities: cos^2(a/2)=(1+cos a)/2, sin^2(a/2)=(1-cos a)/2
        const float ca2 = 0.5f * (1.0f + ca);
        const float sa2 = 0.5f * (1.0f - ca);
        const float sah = 0.5f * sa;

        float fp2r[2], fp2i[2], fm2r[2], fm2i[2];

#pragma unroll
        for (int j = 0; j < 2; ++j) {
            // T2 / T1 relaxation (+ regrowth into Z[0] only)
            float ar = fpr[j] * E2, ai = fpi[j] * E2;
            float br = fmr[j] * E2, bi = fmi[j] * E2;
            float zz = z[j] * E1 + ((j == 0) ? rec0 : 0.0f);

            // B0 precession: Fp *= e^{i phi}, Fm *= e^{-i phi}
            float tpr = ar * b0r - ai * b0i;
            float tpi = ar * b0i + ai * b0r;
            float tmr = br * b0r + bi * b0i;
            float tmi = bi * b0r - br * b0i;

            // RF mixing
            const float saz = sa * zz;
            fp2r[j] = ca2 * tpr + sa2 * (e2r * tmr - e2i * tmi) + saz * ei;
            fp2i[j] = ca2 * tpi + sa2 * (e2r * tmi + e2i * tmr) - saz * er;
            fm2r[j] = sa2 * (e2r * tpr + e2i * tpi) + ca2 * tmr + saz * ei;
            fm2i[j] = sa2 * (e2r * tpi - e2i * tpr) + ca2 * tmi + saz * er;
            z[j] = sah * (er * tpi - ei * tpr - er * tmi - ei * tmr) + ca * zz;
        }

        // State shifts across the wave (wave32 lane permutes):
        //   Fp3[k] = Fp2[k-1]  -> lane L lo gets lane L-1 hi (shuffle up)
        //   Fm3[k] = Fm2[k+1]  -> lane L hi gets lane L+1 lo (shuffle down)
        const float upR = __shfl_up(fp2r[1], 1, 32);
        const float upI = __shfl_up(fp2i[1], 1, 32);
        const float dnR = __shfl_down(fm2r[0], 1, 32);
        const float dnI = __shfl_down(fm2i[0], 1, 32);

        fpr[1] = fp2r[0];  fpi[1] = fp2i[0];
        fpr[0] = upR;      fpi[0] = upI;

        fmr[0] = fm2r[1];  fmi[0] = fm2i[1];
        fmr[1] = (lane == 31) ? 0.0f : dnR;
        fmi[1] = (lane == 31) ? 0.0f : dnI;

        if (lane == 0) {   // Fp3[0] = conj(Fm3[0])
            fpr[0] =  fmr[0];
            fpi[0] = -fmi[0];
        }

        // Streaming stores: 2 complex states -> one b128 NT store per array.
        const size_t cOff = (size_t)p * (BATCH * N_STATES * 2)
                          + (size_t)b * (N_STATES * 2) + (size_t)lane * 4;
        v4f fpv = {fpr[0], fpi[0], fpr[1], fpi[1]};
        v4f fmv = {fmr[0], fmi[0], fmr[1], fmi[1]};
        __builtin_nontemporal_store(fpv, (v4f*)(outFp + cOff));
        __builtin_nontemporal_store(fmv, (v4f*)(outFm + cOff));

        const size_t zOff = (size_t)p * (BATCH * N_STATES)
                          + (size_t)b * N_STATES + (size_t)lane * 2;
        v2f zv = {z[0], z[1]};
        __builtin_nontemporal_store(zv, (v2f*)(outZ + zOff));
    }
}

extern "C" void kernel_launch(void* const* d_in, const int* in_sizes, int n_in,
                              void* d_out, int out_size, void* d_ws, size_t ws_size,
                              hipStream_t stream) {
    (void)in_sizes; (void)n_in; (void)out_size; (void)d_ws; (void)ws_size;
    const float* flip = (const float*)d_in[0];
    const float* phas = (const float*)d_in[1];
    const float* T1   = (const float*)d_in[2];
    const float* T2   = (const float*)d_in[3];
    const int*   TR   = (const int*)  d_in[4];
    // d_in[5] = TE (unused by the reference recurrence)
    const float* B0   = (const float*)d_in[6];
    const float* B1   = (const float*)d_in[7];
    float* out        = (float*)d_out;

    dim3 grid(BATCH / WAVES_PER_BLOCK);   // 256 blocks
    dim3 block(BLOCK_SZ);                 // 8 wave32 per block
    epg_sim_kernel<<<grid, block, 0, stream>>>(flip, phas, T1, T2, TR, B0, B1, out);
}